// DequantAddResidualI8RMSNormQuant_23828478558516
// MI455X (gfx1250) — compile-verified
//
#include <hip/hip_runtime.h>
#include <stdint.h>

typedef float v4f __attribute__((ext_vector_type(4)));
typedef int   v4i __attribute__((ext_vector_type(4)));

#define HIDDEN 4096
#define BLOCK  256          // 8 wave32 per WGP-resident block
#define NWAVES (BLOCK / 32)
#define EPSF   1e-6f

__global__ __launch_bounds__(BLOCK)
void fused_dequant_rmsnorm_quant_kernel(const float* __restrict__ residual,
                                        const int*   __restrict__ x,
                                        const float* __restrict__ weight,
                                        const float* __restrict__ a_ptr,
                                        float* __restrict__ out_res,
                                        float* __restrict__ out_q)
{
    __shared__ float s_w[HIDDEN];      // weight row staged via async DMA
    __shared__ float s_red[NWAVES];    // per-wave sum-of-squares partials

    const int row = blockIdx.x;
    const int tid = threadIdx.x;
    const float a = *a_ptr;

    // ---- CDNA5 async global -> LDS copy of weight (16 KB), overlapped with pass 1 ----
    // the vdst VGPR holds the LDS byte address; generic shared pointer low 32 bits = LDS offset.
    {
        const uint32_t lds_base = (uint32_t)(uintptr_t)&s_w[0];
        const uint64_t gbase    = (uint64_t)(uintptr_t)weight;
#pragma unroll
        for (int i = 0; i < 4; ++i) {
            const uint32_t byteoff = (uint32_t)(i * (BLOCK * 16) + tid * 16);
            uint32_t ldsa = lds_base + byteoff;
            uint64_t ga   = gbase + byteoff;
            asm volatile("global_load_async_to_lds_b128 %0, %1, off"
                         :: "v"(ldsa), "v"(ga) : "memory");
        }
    }

    // ---- pass 1: stream residual + x once (non-temporal), keep res_new in VGPRs ----
    const float* res_row = residual + (size_t)row * HIDDEN;
    const int*   x_row   = x        + (size_t)row * HIDDEN;

    v4f  r[4];
    float ss = 0.0f;
#pragma unroll
    for (int i = 0; i < 4; ++i) {
        const int col = i * 1024 + tid * 4;            // 256 thr * 16 B contiguous per i
        v4f rv = __builtin_nontemporal_load((const v4f*)(res_row + col));
        v4i xv = __builtin_nontemporal_load((const v4i*)(x_row + col));
#pragma unroll
        for (int j = 0; j < 4; ++j) {
            const float t = rv[j] + (float)xv[j] * a;
            r[i][j] = t;
            ss = fmaf(t, t, ss);
        }
    }

    // ---- wave32 reduction (shfl_xor), then cross-wave combine in LDS ----
#pragma unroll
    for (int off = 16; off > 0; off >>= 1)
        ss += __shfl_xor(ss, off, 32);
    if ((tid & 31) == 0) s_red[tid >> 5] = ss;

    // weight DMA must have landed in LDS before pass 2; wave-local wait, then WG barrier
    asm volatile("s_wait_asynccnt 0" ::: "memory");
    __syncthreads();

    float total = 0.0f;
#pragma unroll
    for (int w = 0; w < NWAVES; ++w) total += s_red[w];
    const float inv = rsqrtf(total * (1.0f / (float)HIDDEN) + EPSF);

    // ---- pass 2: write res_new + quantized output (non-temporal), weight from LDS ----
    float* rrow = out_res + (size_t)row * HIDDEN;
    float* qrow = out_q   + (size_t)row * HIDDEN;
#pragma unroll
    for (int i = 0; i < 4; ++i) {
        const int col = i * 1024 + tid * 4;
        const v4f wv = *(const v4f*)&s_w[col];          // ds_load_b128
        const v4f rn = r[i];
        __builtin_nontemporal_store(rn, (v4f*)(rrow + col));
        v4f q;
#pragma unroll
        for (int j = 0; j < 4; ++j) {
            const float y  = rn[j] * inv * wv[j];
            float qq = rintf(y);                        // RNE == jnp.round
            qq = fminf(fmaxf(qq, -128.0f), 127.0f);
            q[j] = qq;
        }
        __builtin_nontemporal_store(q, (v4f*)(qrow + col));
    }
}

extern "C" void kernel_launch(void* const* d_in, const int* in_sizes, int n_in,
                              void* d_out, int out_size, void* d_ws, size_t ws_size,
                              hipStream_t stream) {
    const float* residual = (const float*)d_in[0];
    const int*   x        = (const int*)  d_in[1];
    const float* weight   = (const float*)d_in[2];
    const float* a_ptr    = (const float*)d_in[3];

    const int tokens = in_sizes[0] / HIDDEN;

    float* out_res = (float*)d_out;                         // res_new  [T, H]
    float* out_q   = out_res + (size_t)tokens * HIDDEN;     // quantized [T, H] (as float)

    fused_dequant_rmsnorm_quant_kernel<<<tokens, BLOCK, 0, stream>>>(
        residual, x, weight, a_ptr, out_res, out_q);
}